// GraphNeighborhoodSampler_91242285236448
// MI455X (gfx1250) — compile-verified
//
#include <hip/hip_runtime.h>
#include <hip/hip_bf16.h>

// Sorted k-NN graph sampler for MI455X (gfx1250).
// One workgroup per output row:
//   1) async-stream coords global->LDS (ASYNCcnt double buffer)
//   2) build packed u64 (dist_bits<<32 | idx) keys in 128KB resident LDS
//   3) histogram-select the <=4088 smallest keys (exactly covers top 3001)
//   4) hybrid register/LDS bitonic sort of 4096 keys (16384 fallback)
//   5) emit sorted edges + weights

#define NPAD     16384
#define TILE     2048
#define NTHREADS 1024    // 32 wave32 waves
#define NBINS    2048
#define SELCAP   4088    // 4096 minus 8 control slots

__device__ __forceinline__ unsigned lds_addr_of(const void* p) {
  return (unsigned)(unsigned long long)p;     // low 32 bits = LDS byte address
}
__device__ __forceinline__ void async_copy_b128(unsigned lds_byte_addr, const void* gsrc) {
  asm volatile("global_load_async_to_lds_b128 %0, %1, off"
               :: "v"(lds_byte_addr), "v"(gsrc) : "memory");
}
__device__ __forceinline__ void wait_async_le1() {
  asm volatile("s_wait_asynccnt 1" ::: "memory");
}
__device__ __forceinline__ void wait_async_le0() {
  asm volatile("s_wait_asynccnt 0" ::: "memory");
}

// dx*dx + dy*dy with no FMA contraction: bit-matches jnp sum((a-b)**2, -1).
__device__ __forceinline__ float sq2_rn(float dx, float dy) {
  return __fadd_rn(__fmul_rn(dx, dx), __fmul_rn(dy, dy));
}

__device__ __forceinline__ void cmpswap(unsigned long long& a, unsigned long long& b,
                                        bool up) {
  const bool lt = (a < b);
  const unsigned long long lo = lt ? a : b;
  const unsigned long long hi = lt ? b : a;
  a = up ? lo : hi;
  b = up ? hi : lo;
}

// value -> linear bin; monotone in the key's float part (v >= 0, v < 2.0)
__device__ __forceinline__ int key_bin(unsigned long long key) {
  const float v = __uint_as_float((unsigned)(key >> 32));
  int bin = (int)(v * 1024.0f);
  return (bin > NBINS - 1) ? (NBINS - 1) : bin;
}

// Hybrid register/LDS bitonic sort, ascending, L = power of two >= 16.
__device__ void bitonic_hybrid_sort(unsigned long long* a, unsigned L, int tid) {
  // Stage A: levels ksz = 2,4,8 entirely in registers (aligned 8-key groups)
  for (unsigned g = tid; g < L / 8; g += NTHREADS) {
    const unsigned gbase = g * 8u;
    unsigned long long v[8];
    ulonglong2* gp = (ulonglong2*)&a[gbase];
#pragma unroll
    for (int q = 0; q < 4; ++q) { ulonglong2 t2 = gp[q]; v[2*q] = t2.x; v[2*q+1] = t2.y; }
#pragma unroll
    for (int m = 0; m < 8; m += 2) cmpswap(v[m], v[m + 1], ((m & 2) == 0));
#pragma unroll
    for (int m = 0; m < 8; ++m)
      if ((m & 2) == 0) cmpswap(v[m], v[m | 2], ((m & 4) == 0));
#pragma unroll
    for (int m = 0; m < 8; m += 2) cmpswap(v[m], v[m + 1], ((m & 4) == 0));
    const bool up8 = ((gbase & 8u) == 0);
#pragma unroll
    for (int m = 0; m < 4; ++m) cmpswap(v[m], v[m + 4], up8);
#pragma unroll
    for (int m = 0; m < 8; ++m)
      if ((m & 2) == 0) cmpswap(v[m], v[m | 2], up8);
#pragma unroll
    for (int m = 0; m < 8; m += 2) cmpswap(v[m], v[m + 1], up8);
#pragma unroll
    for (int q = 0; q < 4; ++q) gp[q] = make_ulonglong2(v[2*q], v[2*q+1]);
  }
  __syncthreads();

  // Stage B: levels 16..L; strides j >= 8 in LDS, j = 4,2,1 in registers
  for (unsigned ksz = 16; ksz <= L; ksz <<= 1) {
    for (unsigned j = ksz >> 1; j >= 8; j >>= 1) {
      for (unsigned t2 = tid; t2 < L / 2; t2 += NTHREADS) {
        const unsigned i = ((t2 & ~(j - 1)) << 1) | (t2 & (j - 1));
        const unsigned p = i ^ j;
        const unsigned long long x = a[i];
        const unsigned long long y = a[p];
        const bool up = ((i & ksz) == 0);
        if ((x > y) == up) { a[i] = y; a[p] = x; }
      }
      __syncthreads();
    }
    for (unsigned g = tid; g < L / 8; g += NTHREADS) {
      const unsigned gbase = g * 8u;
      const bool up = ((gbase & ksz) == 0);
      unsigned long long v[8];
      ulonglong2* gp = (ulonglong2*)&a[gbase];
#pragma unroll
      for (int q = 0; q < 4; ++q) { ulonglong2 t2 = gp[q]; v[2*q] = t2.x; v[2*q+1] = t2.y; }
#pragma unroll
      for (int m = 0; m < 4; ++m) cmpswap(v[m], v[m + 4], up);
#pragma unroll
      for (int m = 0; m < 8; ++m)
        if ((m & 2) == 0) cmpswap(v[m], v[m | 2], up);
#pragma unroll
      for (int m = 0; m < 8; m += 2) cmpswap(v[m], v[m + 1], up);
#pragma unroll
      for (int q = 0; q < 4; ++q) gp[q] = make_ulonglong2(v[2*q], v[2*q+1]);
    }
    __syncthreads();
  }
}

__global__ __launch_bounds__(NTHREADS)
void knn_row_kernel(const float2* __restrict__ coords, int n, int K,
                    int* __restrict__ src_o, int* __restrict__ tgt_o,
                    float* __restrict__ w_o, int* __restrict__ k_o) {
  __shared__ float2 s_tile[2][TILE];                 // 32768 B (multi-purpose)
  __shared__ unsigned long long s_key[NPAD];         // 131072 B (total 163840 B)

  const int  b     = blockIdx.x;
  const int  tid   = threadIdx.x;
  const int  kk    = K + 1;
  const bool depot = (b == 0);

  const float tx = coords[b].x;
  const float ty = coords[b].y;

  const int ntiles = (n + TILE - 1) / TILE;

  // ---- phase 1: async-stage tile 0 ----
  {
    const int cnt    = min(TILE, n);
    const int chunks = (cnt + 1) >> 1;
    if (tid < chunks)
      async_copy_b128(lds_addr_of(&s_tile[0][0]) + tid * 16,
                      (const char*)coords + (size_t)tid * 16);
  }
  for (int i = tid; i < NPAD; i += NTHREADS) s_key[i] = ~0ull;  // pad-init

  // ---- phase 2: double-buffered pipeline, build keys ----
  for (int t = 0; t < ntiles; ++t) {
    const int  base     = t * TILE;
    const int  cnt      = min(TILE, n - base);
    const bool has_next = (t + 1 < ntiles);
    if (has_next) {
      const int nbase  = base + TILE;
      const int ncnt   = min(TILE, n - nbase);
      const int chunks = (ncnt + 1) >> 1;
      if (tid < chunks)
        async_copy_b128(lds_addr_of(&s_tile[(t + 1) & 1][0]) + tid * 16,
                        (const char*)coords + (size_t)nbase * 8 + (size_t)tid * 16);
    }
    if (has_next) wait_async_le1(); else wait_async_le0();
    __syncthreads();

    const float2* tl = s_tile[t & 1];
    for (int u = tid; u < cnt; u += NTHREADS) {
      const int j = base + u;
      if (!depot && j == 0) continue;
      const float dx = __fsub_rn(tl[u].x, tx);
      const float dy = __fsub_rn(tl[u].y, ty);
      const float s  = sq2_rn(dx, dy);
      // depot row ranks by norm (reference sorts depot_dists), others by sq
      const float keyval = depot ? ((s > 0.0f) ? __fsqrt_rn(s) : 0.0f) : s;
      const int   slot   = depot ? j : (j - 1);
      s_key[slot] = ((unsigned long long)__float_as_uint(keyval) << 32) |
                    (unsigned)j;
    }
    __syncthreads();
  }

  // ---- phase 3: histogram selection (tile region is now dead -> reuse) ----
  const int Ccount = depot ? n : (n - 1);      // valid keys occupy [0, Ccount)
  unsigned* hA   = (unsigned*)&s_tile[0][0];                   // 2048 u32
  unsigned* hB   = hA + NBINS;                                 // 2048 u32
  unsigned* ctrl = (unsigned*)((char*)&s_tile[0][0] + 32704);  // last 64 B
  unsigned long long* s_sel = (unsigned long long*)&s_tile[0][0]; // 4096 slots

  for (int i = tid; i < NBINS; i += NTHREADS) hA[i] = 0;
  __syncthreads();
  for (int i = tid; i < Ccount; i += NTHREADS)
    atomicAdd(&hA[key_bin(s_key[i])], 1u);
  __syncthreads();

  // inclusive Hillis-Steele scan over 2048 bins (ping-pong hA<->hB)
  unsigned* srcb = hA; unsigned* dstb = hB;
  for (unsigned off = 1; off < NBINS; off <<= 1) {
    for (int i = tid; i < NBINS; i += NTHREADS)
      dstb[i] = srcb[i] + ((i >= (int)off) ? srcb[i - off] : 0u);
    __syncthreads();
    unsigned* tmp = srcb; srcb = dstb; dstb = tmp;
  }

  // smallest bin B with cum(<=B) >= kk  (whole bins <= B cover the kk smallest)
  const unsigned need = (unsigned)kk;
  for (int i = tid; i < NBINS; i += NTHREADS) {
    const unsigned c    = srcb[i];
    const unsigned prev = (i > 0) ? srcb[i - 1] : 0u;
    if (c >= need && prev < need) ctrl[0] = (unsigned)i;
  }
  __syncthreads();
  const unsigned Bsel = ctrl[0];
  const unsigned Csel = srcb[Bsel];
  __syncthreads();                            // everyone has Bsel/Csel; region reusable

  const unsigned long long* out;
  if (Csel <= SELCAP) {
    if (tid == 0) ctrl[1] = 0;                // compaction counter (slot >= SELCAP)
    __syncthreads();
    for (int i = tid; i < Ccount; i += NTHREADS) {
      const unsigned long long key = s_key[i];
      if ((unsigned)key_bin(key) <= Bsel) {
        const unsigned slot = atomicAdd(&ctrl[1], 1u);
        s_sel[slot] = key;
      }
    }
    __syncthreads();
    for (unsigned i = Csel + tid; i < 4096u; i += NTHREADS) s_sel[i] = ~0ull;
    __syncthreads();
    bitonic_hybrid_sort(s_sel, 4096u, tid);
    out = s_sel;
  } else {                                    // degenerate data: full sort fallback
    bitonic_hybrid_sort(s_key, NPAD, tid);
    out = s_key;
  }

  // ---- phase 4: emit row b ----
  const size_t row = (size_t)b * kk;
  for (int c = tid; c < kk; c += NTHREADS) {
    int s_idx, t_idx; float w;
    if (depot) {
      const unsigned long long key = out[c];
      s_idx = (int)(unsigned)key;
      t_idx = 0;
      w     = __uint_as_float((unsigned)(key >> 32));   // key holds the norm
    } else if (c < K) {
      const unsigned long long key = out[c];
      const float s = __uint_as_float((unsigned)(key >> 32));
      s_idx = (int)(unsigned)key;
      t_idx = b;
      w     = (s > 0.0f) ? __fsqrt_rn(s) : 0.0f;
    } else {                                            // trailing depot edge
      s_idx = 0;
      t_idx = b;
      const float dx = __fsub_rn(coords[0].x, tx);
      const float dy = __fsub_rn(coords[0].y, ty);
      const float s  = sq2_rn(dx, dy);
      w     = (s > 0.0f) ? __fsqrt_rn(s) : 0.0f;
    }
    src_o[row + c] = s_idx;
    tgt_o[row + c] = t_idx;
    w_o[row + c]   = w;
  }
  if (depot && tid == 0) *k_o = kk;
}

extern "C" void kernel_launch(void* const* d_in, const int* in_sizes, int n_in,
                              void* d_out, int out_size, void* d_ws, size_t ws_size,
                              hipStream_t stream) {
  (void)n_in; (void)out_size; (void)d_ws; (void)ws_size;
  const float2* coords = (const float2*)d_in[0];
  const int n  = in_sizes[0] / 2;          // (n, 2) float32
  const int K  = (int)(0.3 * (double)n);   // floor(K_FRAC * n) = 3000
  const int kk = K + 1;

  int*   src_o = (int*)d_out;                                  // edge_idx[0]
  int*   tgt_o = src_o + (size_t)n * kk;                       // edge_idx[1]
  float* w_o   = (float*)d_out + 2 * (size_t)n * kk;           // edge_weights
  int*   k_o   = (int*)d_out + 3 * (size_t)n * kk;             // scalar k

  if (n <= 0 || n > NPAD || kk > SELCAP) return;
  knn_row_kernel<<<n, NTHREADS, 0, stream>>>(coords, n, K, src_o, tgt_o, w_o, k_o);
}